// ImageLayer_87737591922785
// MI455X (gfx1250) — compile-verified
//
#include <hip/hip_runtime.h>

// CDNA5 / gfx1250, wave32.
// Persistence-image layer: per (b,p), a 128x128 Gaussian bump.
// Separable Gaussian => rank-1 outer product ey (rows) x ex (cols),
// realized with V_WMMA_F32_16X16X4_F32 (A = ey slice in K=0 column,
// B = ex slice in K=0 row), staged through LDS for coalesced NT stores.
//
// Roofline: output 256 MB @ 23.3 TB/s ~= 11 us; 1M exps + 262k WMMAs are
// negligible -> pure store-bandwidth bound. Design goals: separable trick,
// conflict-free LDS staging, full-row coalesced nontemporal global stores.

typedef __attribute__((ext_vector_type(2))) float v2f;
typedef __attribute__((ext_vector_type(4))) float v4f;
typedef __attribute__((ext_vector_type(8))) float v8f;

#define IMG      128
#define LSTRIDE  132   // LDS row stride (floats): rows stay 16B-aligned, and the
                       // +8-row half-wave offset lands +32 banks -> conflict-free

__global__ __launch_bounds__(256)
void persistence_image_kernel(const float* __restrict__ inp,
                              const float* __restrict__ sg,
                              float* __restrict__ out)
{
    __shared__ float s_ex[IMG];                  // exp term along i0 (columns / N)
    __shared__ float s_ey[IMG];                  // exp term along i1 (rows / M), incl. norm
    __shared__ float s_stage[8][16 * LSTRIDE];   // per-wave 16x128 strip (padded rows)

    const int pair = blockIdx.x;      // b * num_pts + p
    const int tid  = threadIdx.x;     // 0..255 (8 wave32)
    const int wave = tid >> 5;        // 0..7  -> 16-row strip
    const int lane = tid & 31;

    const float s      = sg[0];
    const float inv2s2 = 1.0f / (2.0f * s * s);
    const float norm   = 1.0f / (2.0f * 3.14159265358979f * s * s);

    // birth-persistence transform [[1,-1],[0,1]]:  x stays, y' = y - x
    const float x  = inp[pair * 2 + 0];
    const float yp = inp[pair * 2 + 1] - x;

    const float lo   = -0.0001f;
    const float step = (1.0001f - (-0.0001f)) / 128.0f;

    // ---- Phase 1: 1-D Gaussians (256 threads -> 128 ex + 128 ey) ----
    if (tid < IMG) {
        const float c = lo + step * (float)tid;
        const float d = x - c;
        s_ex[tid] = expf(-d * d * inv2s2);
    } else {
        const int   j = tid - IMG;
        const float c = lo + step * (float)j;
        const float d = yp - c;
        s_ey[j] = norm * expf(-d * d * inv2s2);
    }
    __syncthreads();

    // ---- Phase 2: rank-1 outer product via V_WMMA_F32_16X16X4_F32 ----
    // No divergent branches: all lanes load (16-31 re-read 0-15's address,
    // harmless broadcast) and a float mask zeroes the K=1..3 / upper half.
    const int   l15  = lane & 15;
    const float mask = (lane < 16) ? 1.0f : 0.0f;

    // A (16x4 f32): lanes 0-15 {VGPR0=K0, VGPR1=K1}, lanes 16-31 {K2, K3}.
    // Only the K=0 column is nonzero: A[m][0] = ey[strip + m].
    v2f a;
    a.x = mask * s_ey[wave * 16 + l15];
    a.y = 0.0f;

    // Preload all 8 B slices (row K=0 of each 4x16 B tile) -> one DS wait,
    // then 8 WMMAs issue from registers back-to-back.
    float bx[8];
    #pragma unroll
    for (int tx = 0; tx < 8; ++tx)
        bx[tx] = mask * s_ex[tx * 16 + l15];

    float* stage = &s_stage[wave][0];
    const int mbase = (lane < 16) ? 0 : 8;   // C/D layout: VGPR r -> row mbase + r

    #pragma unroll
    for (int tx = 0; tx < 8; ++tx) {
        v2f b; b.x = bx[tx]; b.y = 0.0f;

        v8f c = {};
        // D = A x B + 0  ==  outer(ey_slice, ex_slice), full fp32.
        c = __builtin_amdgcn_wmma_f32_16x16x4_f32(
                /*neg_a=*/false, a, /*neg_b=*/false, b,
                /*c_mod=*/(short)0, c, /*reuse_a=*/false, /*reuse_b=*/false);

        #pragma unroll
        for (int r = 0; r < 8; ++r)
            stage[(mbase + r) * LSTRIDE + tx * 16 + l15] = c[r];
    }
    __syncthreads();

    // ---- Phase 3: coalesced nontemporal stores. One wave = one 512B row/iter ----
    float* dst = out + (size_t)pair * (IMG * IMG);
    #pragma unroll 4
    for (int r = 0; r < 16; ++r) {
        const int row = wave * 16 + r;
        v4f v = *(const v4f*)&stage[r * LSTRIDE + lane * 4];
        __builtin_nontemporal_store(v, (v4f*)&dst[(size_t)row * IMG + lane * 4]);
    }
}

extern "C" void kernel_launch(void* const* d_in, const int* in_sizes, int n_in,
                              void* d_out, int out_size, void* d_ws, size_t ws_size,
                              hipStream_t stream)
{
    const float* inp = (const float*)d_in[0];   // [B, num_pts, 2] f32
    const float* sg  = (const float*)d_in[1];   // [1] f32
    float* out = (float*)d_out;                 // [B, num_pts, 128*128] f32

    const int pairs = in_sizes[0] / 2;          // B * num_pts = 4096
    persistence_image_kernel<<<pairs, 256, 0, stream>>>(inp, sg, out);
}